// TissuePreservationLoss_43310450213293
// MI455X (gfx1250) — compile-verified
//
#include <hip/hip_runtime.h>
#include <hip/hip_bf16.h>

// Problem geometry: pred/target are [4, 1, 192, 192, 192] fp32.
#define NDIM   192
#define N2     (NDIM * NDIM)            // 36864
#define N3     (NDIM * NDIM * NDIM)     // 7077888
#define NBATCH 4

// Tile: each block owns a 64x8 (x,y) patch and marches the full z range.
#define TX     64
#define TY     8
#define NTHR   (TX * TY)                // 512 threads = 16 wave32
#define PLX    (TX + 2)                 // 66  (x halo)
#define PLY    (TY + 2)                 // 10  (y halo)
#define PLANE  (PLX * PLY)              // 660 floats per halo plane
#define NBUF   4                        // 4-deep rotation -> DMA/compute overlap
// LDS: 4 rotating z-planes x {pred,target} + 16-float reduction scratch
#define SMEM_F (2 * NBUF * PLANE + 16)

#define GRIDX  (NDIM / TX)              // 3
#define GRIDY  (NDIM / TY)              // 24
#define NBLK   (GRIDX * GRIDY * NBATCH) // 288 partial sums

// ---------------------------------------------------------------------------
// CDNA5 async global->LDS copy (ASYNCcnt-tracked DMA, no VGPR staging).
// clang signature: (int addrspace(1)*, int addrspace(3)*, imm offset, imm cpol)
// ---------------------------------------------------------------------------
#if defined(__HIP_DEVICE_COMPILE__) && __has_builtin(__builtin_amdgcn_global_load_async_to_lds_b32)
#define HAVE_ASYNC_BUILTIN 1
#endif

typedef __attribute__((address_space(1))) int  g_int;
typedef __attribute__((address_space(3))) int  l_int;

__device__ __forceinline__ void async_cp_f32(const float* g, float* smem, int idx) {
#if defined(HAVE_ASYNC_BUILTIN)
    __builtin_amdgcn_global_load_async_to_lds_b32(
        (g_int*)g, (l_int*)(smem + idx), 0, 0);
#else
    // Fallback: the single __shared__ block starts at LDS offset 0,
    // so the LDS byte address is just idx*4.
    (void)smem;
    asm volatile("global_load_async_to_lds_b32 %0, %1, off"
                 :: "v"((unsigned)(idx * 4)), "v"(g)
                 : "memory");
#endif
}

__device__ __forceinline__ void wait_async0() {
#if defined(__HIP_DEVICE_COMPILE__) && __has_builtin(__builtin_amdgcn_s_wait_asynccnt)
    __builtin_amdgcn_s_wait_asynccnt(0);
#else
    asm volatile("s_wait_asynccnt 0" ::: "memory");
#endif
}

// Issue async loads of one z-plane (with x/y halo) of pred+target into LDS
// buffer `buf`. Out-of-range halo cells are simply not written: they were
// zeroed once at kernel start and stay zero -> implements zero padding.
__device__ __forceinline__ void issue_plane(const float* __restrict__ pred,
                                            const float* __restrict__ targ,
                                            float* smem, size_t bbase, int z,
                                            int x0, int y0, int buf, int tid) {
    const size_t zbase = bbase + (size_t)z * (size_t)N2;
#pragma unroll
    for (int idx = tid; idx < PLANE; idx += NTHR) {
        const int ly = idx / PLX;
        const int lx = idx - ly * PLX;
        const int gy = y0 + ly - 1;
        const int gx = x0 + lx - 1;
        if ((unsigned)gy < (unsigned)NDIM && (unsigned)gx < (unsigned)NDIM) {
            const size_t goff = zbase + (size_t)gy * (size_t)NDIM + (size_t)gx;
            async_cp_f32(pred + goff, smem, (buf * 2 + 0) * PLANE + idx);
            async_cp_f32(targ + goff, smem, (buf * 2 + 1) * PLANE + idx);
        }
    }
}

// ---------------------------------------------------------------------------
// Pass 1: per-block partial sum of (|lap(pred)| - |lap(target)|)^2
//
// 4-buffer software pipeline: plane p lives in buffer (p & 3).
// Iteration z: issue DMA for plane z+2 into buf (z+2)&3, then compute from
// planes z-1, z, z+1 (buffers (z+3)&3, z&3, (z+1)&3 -- disjoint from the DMA
// target), then s_wait_asynccnt + one barrier. DMA overlaps compute.
// ---------------------------------------------------------------------------
__global__ __launch_bounds__(NTHR) void lap_loss_partial(
    const float* __restrict__ pred, const float* __restrict__ targ,
    float* __restrict__ partials) {
    __shared__ float smem[SMEM_F];

    const int tx  = threadIdx.x;
    const int ty  = threadIdx.y;
    const int tid = ty * TX + tx;
    const int x0  = (int)blockIdx.x * TX;
    const int y0  = (int)blockIdx.y * TY;
    const size_t bbase = (size_t)blockIdx.z * (size_t)N3;

    // Zero all of LDS once: provides the z=-1 plane (buf 3) and permanent
    // zero padding for out-of-volume halo cells (never async-written).
    for (int i = tid; i < SMEM_F; i += NTHR) smem[i] = 0.0f;
    __syncthreads();   // DS stores complete before async DMA can land

    // Prime: plane 0 -> buf0, plane 1 -> buf1 (buf3 = zeros = plane -1).
    issue_plane(pred, targ, smem, bbase, 0, x0, y0, 0, tid);
    issue_plane(pred, targ, smem, bbase, 1, x0, y0, 1, tid);
    wait_async0();
    __syncthreads();

    const int cen = (ty + 1) * PLX + (tx + 1);
    float acc = 0.0f;

    for (int z = 0; z < NDIM; ++z) {
        const int bl = (z + 2) & 3;           // buffer receiving plane z+2
        if (z < NDIM - 2) {
            // Prefetch plane z+2 while we compute plane z.
            issue_plane(pred, targ, smem, bbase, z + 2, x0, y0, bl, tid);
        } else if (z == NDIM - 2) {
            // plane 192 == zeros; only center cells are ever read from bp
            smem[(bl * 2 + 0) * PLANE + cen] = 0.0f;
            smem[(bl * 2 + 1) * PLANE + cen] = 0.0f;
        }

        const int pm = (((z + 3) & 3) * 2) * PLANE;  // plane z-1 (center only)
        const int pc = (((z    ) & 3) * 2) * PLANE;  // plane z   (x/y halo)
        const int pp = (((z + 1) & 3) * 2) * PLANE;  // plane z+1 (center only)

        // pred Laplacian
        const float c0 = smem[pc + cen];
        const float lp = smem[pc + cen - 1] + smem[pc + cen + 1]
                       + smem[pc + cen - PLX] + smem[pc + cen + PLX]
                       + smem[pm + cen] + smem[pp + cen]
                       - 6.0f * c0;
        // target Laplacian (pred/target interleaved per buffer, +PLANE)
        const float c1 = smem[pc + PLANE + cen];
        const float lt = smem[pc + PLANE + cen - 1] + smem[pc + PLANE + cen + 1]
                       + smem[pc + PLANE + cen - PLX] + smem[pc + PLANE + cen + PLX]
                       + smem[pm + PLANE + cen] + smem[pp + PLANE + cen]
                       - 6.0f * c1;

        const float d = fabsf(lp) - fabsf(lt);
        acc = fmaf(d, d, acc);

        // Single barrier per plane: makes plane z+2 visible for the next
        // iteration AND retires all reads of the buffer that iteration z+1
        // will overwrite.
        wait_async0();
        __syncthreads();
    }

    // Deterministic block reduction: wave32 shuffle, then cross-wave via LDS.
    for (int off = 16; off > 0; off >>= 1) acc += __shfl_down(acc, off, 32);
    const int lane = tid & 31;
    const int wave = tid >> 5;
    if (lane == 0) smem[2 * NBUF * PLANE + wave] = acc;
    __syncthreads();
    if (tid < 32) {
        float v = (tid < (NTHR / 32)) ? smem[2 * NBUF * PLANE + tid] : 0.0f;
        for (int off = 8; off > 0; off >>= 1) v += __shfl_down(v, off, 32);
        if (tid == 0) {
            const int pidx = ((int)blockIdx.z * GRIDY + (int)blockIdx.y) * GRIDX
                             + (int)blockIdx.x;
            partials[pidx] = v;
        }
    }
}

// ---------------------------------------------------------------------------
// Pass 2: deterministic final mean over the 288 block partials.
// ---------------------------------------------------------------------------
__global__ __launch_bounds__(256) void reduce_mean(
    const float* __restrict__ parts, float* __restrict__ out) {
    __shared__ float red[8];
    const int tid = threadIdx.x;
    float s = 0.0f;
    for (int i = tid; i < NBLK; i += 256) s += parts[i];
    for (int off = 16; off > 0; off >>= 1) s += __shfl_down(s, off, 32);
    if ((tid & 31) == 0) red[tid >> 5] = s;
    __syncthreads();
    if (tid < 32) {
        float v = (tid < 8) ? red[tid] : 0.0f;
        for (int off = 4; off > 0; off >>= 1) v += __shfl_down(v, off, 32);
        if (tid == 0)
            out[0] = v * (1.0f / (float)((size_t)NBATCH * (size_t)N3));
    }
}

// ---------------------------------------------------------------------------
extern "C" void kernel_launch(void* const* d_in, const int* in_sizes, int n_in,
                              void* d_out, int out_size, void* d_ws, size_t ws_size,
                              hipStream_t stream) {
    (void)in_sizes; (void)n_in; (void)out_size; (void)ws_size;
    const float* pred = (const float*)d_in[0];
    const float* targ = (const float*)d_in[1];
    float* out   = (float*)d_out;
    float* parts = (float*)d_ws;    // 288 floats of scratch

    dim3 grid(GRIDX, GRIDY, NBATCH);
    dim3 block(TX, TY, 1);
    hipLaunchKernelGGL(lap_loss_partial, grid, block, 0, stream, pred, targ, parts);
    hipLaunchKernelGGL(reduce_mean, dim3(1), dim3(256), 0, stream, parts, out);
}